// ReadHead_3204045603566
// MI455X (gfx1250) — compile-verified
//
#include <hip/hip_runtime.h>

#define HDIM  64
#define BATCH 128
#define MEMN  32768
#define SPLITS 8
#define ROWS_PER_BLOCK (MEMN / SPLITS)          // 4096
#define WAVES_PER_BLOCK 8
#define ROWS_PER_WAVE (ROWS_PER_BLOCK / WAVES_PER_BLOCK)  // 512
#define PSTRIDE (HDIM + 2)                      // per-partial: {max, sum, acc[64]}

typedef __attribute__((ext_vector_type(2))) float v2f;
typedef __attribute__((ext_vector_type(8))) float v8f;

// ---------------------------------------------------------------------------
// D[128,64] = A[128,64] @ W[64,64]^T + bias[64], fp32 via V_WMMA_F32_16X16X4_F32
// Grid: 32 blocks (8 row-tiles x 4 col-tiles), 32 threads (one wave per tile).
// ---------------------------------------------------------------------------
__global__ __launch_bounds__(32) void wmma_proj64(
    const float* __restrict__ A, const float* __restrict__ W,
    const float* __restrict__ bias, float* __restrict__ D) {
  const int tile = blockIdx.x;
  const int tm = tile >> 2;            // 0..7  (16-row tile of A/D)
  const int tn = tile & 3;             // 0..3  (16-col tile of D = 16 rows of W)
  const int lane = threadIdx.x;        // 0..31 (wave32)
  const int n = lane & 15;             // M for A-frag, N for B/C/D-frags
  const int hi = lane >> 4;            // lane-half selects K pair / M+8

  // C/D layout: VGPR r on lane l -> D[tm*16 + r + 8*hi][tn*16 + (l&15)]
  v8f c;
  const float bval = bias[tn * 16 + n];
#pragma unroll
  for (int r = 0; r < 8; ++r) c[r] = bval;

  const float* arow = A + (size_t)(tm * 16 + n) * HDIM;  // A[m][*]
  const float* wrow = W + (size_t)(tn * 16 + n) * HDIM;  // W[n][*] == B^T row

#pragma unroll
  for (int kk = 0; kk < 16; ++kk) {
    // f32 16x4 A-frag: lanes 0-15 -> K = {0,1}, lanes 16-31 -> K = {2,3}
    const int k0 = kk * 4 + hi * 2;
    v2f a = *(const v2f*)(arow + k0);  // A[m][k0], A[m][k0+1]
    v2f b = *(const v2f*)(wrow + k0);  // B[k0][n], B[k0+1][n] (= W[n][k])
    c = __builtin_amdgcn_wmma_f32_16x16x4_f32(
        /*neg_a=*/false, a, /*neg_b=*/false, b,
        /*c_mod=*/(short)0, c, /*reuse_a=*/false, /*reuse_b=*/false);
  }

#pragma unroll
  for (int r = 0; r < 8; ++r)
    D[(size_t)(tm * 16 + r + hi * 8) * HDIM + tn * 16 + n] = c[r];
}

// ---------------------------------------------------------------------------
// Fused scores + online softmax + weighted pooling, single pass over memory.
// Grid: BATCH*SPLITS blocks of 256 threads (8 waves). Each lane owns 2 columns
// -> one wave streams two full 256B rows per iteration as coalesced NT b64 loads.
// Emits per-(batch,split) partial {running_max, running_sum, acc[64]}.
// ---------------------------------------------------------------------------
__global__ __launch_bounds__(256) void attn_pool_partial(
    const float* __restrict__ memory, const float* __restrict__ q,
    float* __restrict__ partials) {
  const int blk = blockIdx.x;
  const int b = blk / SPLITS;
  const int s = blk % SPLITS;
  const int tid = threadIdx.x;
  const int wave = tid >> 5;
  const int lane = tid & 31;

  const v2f qv = *(const v2f*)(q + (size_t)b * HDIM + lane * 2);

  const int row0 = s * ROWS_PER_BLOCK + wave * ROWS_PER_WAVE;
  const float* mp =
      memory + (size_t)b * MEMN * HDIM + (size_t)row0 * HDIM + lane * 2;

  float accx = 0.f, accy = 0.f;
  float mrun = -__builtin_inff(), srun = 0.f;

  for (int r = 0; r < ROWS_PER_WAVE; r += 2) {
    // Two rows in flight: streaming, zero-reuse -> non-temporal loads.
    v2f mv0 = __builtin_nontemporal_load((const v2f*)(mp));
    v2f mv1 = __builtin_nontemporal_load((const v2f*)(mp + HDIM));
    float p0 = mv0.x * qv.x + mv0.y * qv.y;
    float p1 = mv1.x * qv.x + mv1.y * qv.y;
#pragma unroll
    for (int off = 16; off > 0; off >>= 1) {
      p0 += __shfl_xor(p0, off, 32);
      p1 += __shfl_xor(p1, off, 32);
    }
    // Online softmax updates (wave-uniform d -> uniform branch).
    {
      float d = p0 * 0.125f;  // 1/sqrt(64)
      if (d > mrun) {
        float f = __expf(mrun - d);
        srun *= f; accx *= f; accy *= f; mrun = d;
      }
      float w = __expf(d - mrun);
      srun += w; accx += w * mv0.x; accy += w * mv0.y;
    }
    {
      float d = p1 * 0.125f;
      if (d > mrun) {
        float f = __expf(mrun - d);
        srun *= f; accx *= f; accy *= f; mrun = d;
      }
      float w = __expf(d - mrun);
      srun += w; accx += w * mv1.x; accy += w * mv1.y;
    }
    mp += 2 * HDIM;
  }

  // Merge the 8 wave-partials via LDS.
  __shared__ float sm[WAVES_PER_BLOCK];
  __shared__ float ss[WAVES_PER_BLOCK];
  __shared__ float sa[WAVES_PER_BLOCK][HDIM];
  if (lane == 0) { sm[wave] = mrun; ss[wave] = srun; }
  sa[wave][lane * 2]     = accx;
  sa[wave][lane * 2 + 1] = accy;
  __syncthreads();

  if (tid < HDIM) {
    float mmax = sm[0];
#pragma unroll
    for (int i = 1; i < WAVES_PER_BLOCK; ++i) mmax = fmaxf(mmax, sm[i]);
    float stot = 0.f, a = 0.f;
#pragma unroll
    for (int i = 0; i < WAVES_PER_BLOCK; ++i) {
      float f = __expf(sm[i] - mmax);
      stot += ss[i] * f;
      a += sa[i][tid] * f;
    }
    float* P = partials + (size_t)blk * PSTRIDE;
    P[2 + tid] = a;
    if (tid == 0) { P[0] = mmax; P[1] = stot; }
  }
}

// ---------------------------------------------------------------------------
// Merge SPLITS partials per batch, normalize -> pooled[128,64].
// ---------------------------------------------------------------------------
__global__ __launch_bounds__(64) void finalize_pool(
    const float* __restrict__ partials, float* __restrict__ pooled) {
  const int b = blockIdx.x;   // 128
  const int c = threadIdx.x;  // 64
  const float* P = partials + (size_t)b * SPLITS * PSTRIDE;
  float mmax = P[0];
#pragma unroll
  for (int s = 1; s < SPLITS; ++s) mmax = fmaxf(mmax, P[s * PSTRIDE]);
  float stot = 0.f, a = 0.f;
#pragma unroll
  for (int s = 0; s < SPLITS; ++s) {
    const float* Ps = P + s * PSTRIDE;
    float f = __expf(Ps[0] - mmax);
    stot += Ps[1] * f;
    a += Ps[2 + c] * f;
  }
  pooled[(size_t)b * HDIM + c] = a / stot;
}

// ---------------------------------------------------------------------------
extern "C" void kernel_launch(void* const* d_in, const int* in_sizes, int n_in,
                              void* d_out, int out_size, void* d_ws,
                              size_t ws_size, hipStream_t stream) {
  const float* query_h = (const float*)d_in[0];  // [128,64]
  const float* memory  = (const float*)d_in[1];  // [128,32768,64]
  const float* q_w     = (const float*)d_in[2];  // [64,64]
  const float* q_b     = (const float*)d_in[3];  // [64]
  const float* out_w   = (const float*)d_in[4];  // [64,64]
  const float* out_b   = (const float*)d_in[5];  // [64]
  float* out = (float*)d_out;                    // [128,64]

  float* ws = (float*)d_ws;
  float* Q  = ws;                                   // 128*64
  float* P  = Q + BATCH * HDIM;                     // 128*8*66
  float* PL = P + BATCH * SPLITS * PSTRIDE;         // 128*64

  // q = query_h @ q_w.T + q_b   (fp32 WMMA)
  wmma_proj64<<<32, 32, 0, stream>>>(query_h, q_w, q_b, Q);
  // fused scores/softmax/pool, single 1 GiB NT stream of `memory`
  attn_pool_partial<<<BATCH * SPLITS, 256, 0, stream>>>(memory, Q, P);
  finalize_pool<<<BATCH, HDIM, 0, stream>>>(P, PL);
  // out = pooled @ out_w.T + out_b  (fp32 WMMA)
  wmma_proj64<<<32, 32, 0, stream>>>(PL, out_w, out_b, out);
}